// RoIMaskAlign_8538394984626
// MI455X (gfx1250) — compile-verified
//
#include <hip/hip_runtime.h>

// RoI Mask Align for MI455X (gfx1250).
// Memory-bound gather kernel: one block per RoI, thread = output bin,
// channel loop with register-resident bilinear weights/offsets and
// coalesced output stores. The per-RoI 64x64 bilinear weight matrix
// (wy outer-product wx, with validity mask and 1/(SR*SR) folded in) is
// built with V_WMMA_F32_16X16X4_F32 in LDS.

#define BDIM   2
#define CDIM   256
#define HDIM   200
#define WDIM   272
#define PH_    14
#define PW_    14
#define NBINS  (PH_ * PW_)          // 196
#define SCALE  0.25f
#define RSCALE 1.2f

typedef float v2f __attribute__((ext_vector_type(2)));
typedef float v8f __attribute__((ext_vector_type(8)));

__global__ __launch_bounds__(256) void roi_mask_align_kernel(
    const float* __restrict__ feat,   // (B, C, H, W)
    const float* __restrict__ rois,   // (K, 5)
    float* __restrict__ out)          // (K, C, PH, PW)
{
    __shared__ float s_wy[64];
    __shared__ float s_wx[64];
    __shared__ float s_W[64 * 64];    // 16 KB weight matrix
    __shared__ int   s_y0[28], s_y1[28], s_x0[28], s_x1[28];

    const int roi = blockIdx.x;
    const int tid = threadIdx.x;

    if (tid < 64) { s_wy[tid] = 0.0f; s_wx[tid] = 0.0f; }
    __syncthreads();

    // ---- RoI transform (uniform across block; scalar-cached loads) ----
    const float rb  = rois[roi * 5 + 0];
    const float rx1 = rois[roi * 5 + 1];
    const float ry1 = rois[roi * 5 + 2];
    const float rx2 = rois[roi * 5 + 3];
    const float ry2 = rois[roi * 5 + 4];
    const int   bi  = (int)rb;

    const float wbox = (rx2 - rx1) * RSCALE;
    const float hbox = (ry2 - ry1) * RSCALE;
    const float x1f  = ((rx1 + rx2) * 0.5f - 0.5f * wbox) * SCALE;
    const float y1f  = ((ry1 + ry2) * 0.5f - 0.5f * hbox) * SCALE;
    const float roiw = fmaxf(wbox * SCALE, 1.0f);
    const float roih = fmaxf(hbox * SCALE, 1.0f);
    const float binw = roiw * (1.0f / (float)PW_);
    const float binh = roih * (1.0f / (float)PH_);

    // ---- per-axis sample coords, clamp, validity; 0.5 fold per axis ----
    if (tid < 28) {                       // y axis: (ph, sy)
        const int   ph = tid >> 1, sy = tid & 1;
        const float fr = sy ? 0.75f : 0.25f;
        const float y  = y1f + ((float)ph + fr) * binh;
        const bool  v  = (y > -1.0f) && (y < (float)HDIM);
        const float yc = fminf(fmaxf(y, 0.0f), (float)(HDIM - 1));
        const float yf = floorf(yc);
        const int   i0 = (int)yf;
        const int   i1 = min(i0 + 1, HDIM - 1);
        const float ly = yc - yf;
        s_y0[tid] = i0;
        s_y1[tid] = i1;
        s_wy[ph * 4 + sy * 2 + 0] = v ? (1.0f - ly) * 0.5f : 0.0f;
        s_wy[ph * 4 + sy * 2 + 1] = v ? ly * 0.5f : 0.0f;
    } else if (tid >= 32 && tid < 60) {   // x axis: (pw, sx)
        const int   j  = tid - 32;
        const int   pw = j >> 1, sx = j & 1;
        const float fr = sx ? 0.75f : 0.25f;
        const float x  = x1f + ((float)pw + fr) * binw;
        const bool  v  = (x > -1.0f) && (x < (float)WDIM);
        const float xc = fminf(fmaxf(x, 0.0f), (float)(WDIM - 1));
        const float xf = floorf(xc);
        const int   i0 = (int)xf;
        const int   i1 = min(i0 + 1, WDIM - 1);
        const float lx = xc - xf;
        s_x0[j] = i0;
        s_x1[j] = i1;
        s_wx[pw * 4 + sx * 2 + 0] = v ? (1.0f - lx) * 0.5f : 0.0f;
        s_wx[pw * 4 + sx * 2 + 1] = v ? lx * 0.5f : 0.0f;
    }
    __syncthreads();

    // ---- W = wy (64) outer wx (64) via V_WMMA_F32_16X16X4_F32 ----
    // A 16x4 f32: lanes 0-15 hold {K=0,K=1}, lanes 16-31 hold {K=2,K=3}.
    // Only K=0 carries data (wy / wx chunks); other K slices are zero.
#if __has_builtin(__builtin_amdgcn_wmma_f32_16x16x4_f32)
    {
        const int wave = tid >> 5;
        const int lane = tid & 31;
        if (wave < 4) {                   // 4 waves x 4 tiles = 16 tiles
#pragma unroll
            for (int k = 0; k < 4; ++k) {
                const int tile = wave * 4 + k;
                const int ti = tile >> 2;     // row tile
                const int tj = tile & 3;      // col tile
                v2f a, b;
                a.x = (lane < 16) ? s_wy[ti * 16 + lane] : 0.0f;
                a.y = 0.0f;
                b.x = (lane < 16) ? s_wx[tj * 16 + lane] : 0.0f;
                b.y = 0.0f;
                v8f c = {};
                c = __builtin_amdgcn_wmma_f32_16x16x4_f32(
                        false, a, false, b, (short)0, c, false, false);
                // D 16x16 f32: M = r + 8*(lane>=16), N = lane & 15
                const int row0 = ti * 16 + ((lane >= 16) ? 8 : 0);
                const int col  = tj * 16 + (lane & 15);
#pragma unroll
                for (int r = 0; r < 8; ++r)
                    s_W[(row0 + r) * 64 + col] = c[r];
            }
        }
    }
#else
    for (int i = tid; i < 64 * 64; i += 256)
        s_W[i] = s_wy[i >> 6] * s_wx[i & 63];
#endif
    __syncthreads();

    // ---- main gather: thread = bin, loop over channels ----
    if (tid < NBINS) {
        const int ph = tid / PW_;
        const int pw = tid % PW_;

        float wreg[16];
        int   offr[16];
#pragma unroll
        for (int s = 0; s < 16; ++s) {
            const int sy = (s >> 3) & 1;
            const int cy = (s >> 2) & 1;
            const int sx = (s >> 1) & 1;
            const int cx =  s       & 1;
            wreg[s] = s_W[(ph * 4 + sy * 2 + cy) * 64 + (pw * 4 + sx * 2 + cx)];
            const int yy = cy ? s_y1[ph * 2 + sy] : s_y0[ph * 2 + sy];
            const int xx = cx ? s_x1[pw * 2 + sx] : s_x0[pw * 2 + sx];
            offr[s] = yy * WDIM + xx;
        }

        const size_t plane = (size_t)HDIM * WDIM;
        const float* __restrict__ base = feat + (size_t)bi * CDIM * plane;
        float* __restrict__ obase = out + (size_t)roi * CDIM * NBINS + tid;

        // 2-channel unroll: 32 outstanding gathers per iteration.
        for (int c = 0; c < CDIM; c += 2) {
            const float* __restrict__ p0 = base + (size_t)c * plane;
            const float* __restrict__ p1 = p0 + plane;
            float a0 = 0.0f, a1 = 0.0f;
#pragma unroll
            for (int s = 0; s < 16; ++s) {
                a0 += wreg[s] * p0[offr[s]];
                a1 += wreg[s] * p1[offr[s]];
            }
            // coalesced: 196 lanes write 196 consecutive floats
            obase[(size_t)c * NBINS]       = a0;
            obase[(size_t)(c + 1) * NBINS] = a1;
        }
    }
}

extern "C" void kernel_launch(void* const* d_in, const int* in_sizes, int n_in,
                              void* d_out, int out_size, void* d_ws, size_t ws_size,
                              hipStream_t stream) {
    (void)n_in; (void)out_size; (void)d_ws; (void)ws_size;
    const float* feat = (const float*)d_in[0];
    const float* rois = (const float*)d_in[1];
    float* out = (float*)d_out;
    const int K = in_sizes[1] / 5;     // 256 RoIs
    roi_mask_align_kernel<<<K, 256, 0, stream>>>(feat, rois, out);
}